// ModelNew_23828478558513
// MI455X (gfx1250) — compile-verified
//
#include <hip/hip_runtime.h>
#include <hip/hip_bf16.h>

// Mamba selective scan, chunked parallel formulation for MI455X (gfx1250).
//  Pass A: per (b, d, chunk): chunk decay product P[n]=exp(A[n]*sum(dt)) and
//          local state xloc[n] (scan from x=0)
//  Pass B: per (b, d, n): scan over chunks -> chunk-initial states (overwrites xloc ws)
//  Pass C: per (b, d, chunk): replay chunk with correct initial state, emit y
// CDNA5 features: async global->LDS B128 staging of B/C tiles (ASYNCcnt +
// s_wait_asynccnt), global_prefetch hints, wave32-native 256-thread blocks.

#define BATCH 4
#define DIMS  1536
#define SEQ   4096
#define NST   16
#define CHUNK 64
#define NCH   (SEQ / CHUNK)
#define DTILE 256
#define NDT   (DIMS / DTILE)

#if defined(__AMDGCN__) && __has_builtin(__builtin_amdgcn_global_load_async_to_lds_b128)
#define HAVE_ASYNC_LDS 1
typedef int v4i_t __attribute__((vector_size(16)));
typedef __attribute__((address_space(1))) v4i_t gv4i_t;  // global 16B vector
typedef __attribute__((address_space(3))) v4i_t lv4i_t;  // LDS 16B vector
#else
#define HAVE_ASYNC_LDS 0
#endif

__device__ __forceinline__ void async_copy16(const float* g, float* l) {
#if HAVE_ASYNC_LDS
  // per-lane 16B global -> LDS DMA, tracked with ASYNCcnt
  __builtin_amdgcn_global_load_async_to_lds_b128((gv4i_t*)g, (lv4i_t*)l, 0, 0);
#else
  *(float4*)l = *(const float4*)g;
#endif
}

__device__ __forceinline__ void async_wait_all() {
#if HAVE_ASYNC_LDS
#if __has_builtin(__builtin_amdgcn_s_wait_asynccnt)
  __builtin_amdgcn_s_wait_asynccnt(0);
#else
  asm volatile("s_wait_asynccnt 0x0" ::: "memory");
#endif
#endif
}

__device__ __forceinline__ float softplus_f(float v) {
  // jax.nn.softplus = log1p(exp(v)); large-v branch avoids overflow
  return (v > 20.0f) ? v : __logf(1.0f + __expf(v));
}

// ---------------- Pass A: per-chunk local scan ----------------
__global__ __launch_bounds__(256) void ssm_chunk_local(
    const float* __restrict__ u, const float* __restrict__ delta,
    const float* __restrict__ Bm, const float* __restrict__ A,
    const float* __restrict__ delta_bias,
    float* __restrict__ aprod_ws, float* __restrict__ xloc_ws) {
  __shared__ float sB[NST * CHUNK];

  const int tid = threadIdx.x;
  const int dtile = blockIdx.x, c = blockIdx.y, b = blockIdx.z;
  const int d = dtile * DTILE + tid;
  const int s0 = c * CHUNK;

  // Stage B[b, :, s0:s0+64] (16x64 f32 = 4KB): one async B128 per lane.
  {
    const float* g = Bm + (size_t)(b * NST + (tid >> 4)) * SEQ + s0 + ((tid & 15) << 2);
    async_copy16(g, &sB[tid << 2]);
    async_wait_all();
  }
  __syncthreads();

  float Arow[NST];
#pragma unroll
  for (int n = 0; n < NST; n += 4) {
    float4 t = *(const float4*)&A[(size_t)d * NST + n];
    Arow[n] = t.x; Arow[n + 1] = t.y; Arow[n + 2] = t.z; Arow[n + 3] = t.w;
  }
  const float bias = delta_bias[d];

  float x[NST];
#pragma unroll
  for (int n = 0; n < NST; n++) x[n] = 0.0f;
  float dtsum = 0.0f;  // prod_t exp(dt_t*A[n]) == exp(A[n]*sum_t dt_t)

  const float* up = u + (size_t)(b * DIMS + d) * SEQ + s0;
  const float* dp = delta + (size_t)(b * DIMS + d) * SEQ + s0;

  // warm the next chunk's lines in L2 (speculative, dropped if OOB)
  __builtin_prefetch(up + CHUNK, 0, 1);
  __builtin_prefetch(dp + CHUNK, 0, 1);

  for (int j = 0; j < CHUNK; j += 4) {
    float4 u4 = *(const float4*)(up + j);
    float4 l4 = *(const float4*)(dp + j);
#pragma unroll
    for (int k = 0; k < 4; k++) {
      float dt = softplus_f(((const float*)&l4)[k] + bias);
      float du = dt * ((const float*)&u4)[k];
      dtsum += dt;
      const float* bj = &sB[j + k];
#pragma unroll
      for (int n = 0; n < NST; n++) {
        float a = __expf(dt * Arow[n]);
        x[n] = a * x[n] + du * bj[n * CHUNK];  // LDS broadcast read
      }
    }
  }

  float* apo = aprod_ws + (size_t)((b * NCH + c) * DIMS + d) * NST;
  float* xlo = xloc_ws  + (size_t)((b * NCH + c) * DIMS + d) * NST;
#pragma unroll
  for (int n = 0; n < NST; n += 4) {
    *(float4*)(apo + n) = make_float4(__expf(dtsum * Arow[n]),
                                      __expf(dtsum * Arow[n + 1]),
                                      __expf(dtsum * Arow[n + 2]),
                                      __expf(dtsum * Arow[n + 3]));
    *(float4*)(xlo + n) = make_float4(x[n], x[n + 1], x[n + 2], x[n + 3]);
  }
}

// ---------------- Pass B: scan over chunks ----------------
__global__ __launch_bounds__(256) void ssm_chunk_scan(
    const float* __restrict__ aprod_ws, float* __restrict__ xloc_ws) {
  const int idx = blockIdx.x * 256 + threadIdx.x;  // flat (b, d, n)
  if (idx >= BATCH * DIMS * NST) return;
  const int n = idx & (NST - 1);
  const int bd = idx / NST;
  const int b = bd / DIMS, d = bd % DIMS;

  float x = 0.0f;
  for (int c = 0; c < NCH; c++) {
    const size_t a = (size_t)((b * NCH + c) * DIMS + d) * NST + n;
    const float apv = aprod_ws[a];
    const float xlv = xloc_ws[a];
    xloc_ws[a] = x;          // chunk-initial state (consumed by pass C)
    x = apv * x + xlv;       // combine: (P1,x1) o (P2,x2)
  }
}

// ---------------- Pass C: replay chunk with init state, emit y ----------------
__global__ __launch_bounds__(256) void ssm_chunk_out(
    const float* __restrict__ u, const float* __restrict__ delta,
    const float* __restrict__ Bm, const float* __restrict__ Cm,
    const float* __restrict__ A, const float* __restrict__ Dv,
    const float* __restrict__ delta_bias,
    const float* __restrict__ xinit_ws, float* __restrict__ y) {
  __shared__ float sB[NST * CHUNK];
  __shared__ float sC[NST * CHUNK];

  const int tid = threadIdx.x;
  const int dtile = blockIdx.x, c = blockIdx.y, b = blockIdx.z;
  const int d = dtile * DTILE + tid;
  const int s0 = c * CHUNK;

  {
    const size_t off = (size_t)(b * NST + (tid >> 4)) * SEQ + s0 + ((tid & 15) << 2);
    async_copy16(Bm + off, &sB[tid << 2]);
    async_copy16(Cm + off, &sC[tid << 2]);
    async_wait_all();
  }
  __syncthreads();

  float Arow[NST], x[NST];
  const float* xin = xinit_ws + (size_t)((b * NCH + c) * DIMS + d) * NST;
#pragma unroll
  for (int n = 0; n < NST; n += 4) {
    float4 t = *(const float4*)&A[(size_t)d * NST + n];
    Arow[n] = t.x; Arow[n + 1] = t.y; Arow[n + 2] = t.z; Arow[n + 3] = t.w;
    float4 xi = *(const float4*)(xin + n);
    x[n] = xi.x; x[n + 1] = xi.y; x[n + 2] = xi.z; x[n + 3] = xi.w;
  }
  const float bias = delta_bias[d];
  const float Dd = Dv[d];

  const float* up = u + (size_t)(b * DIMS + d) * SEQ + s0;
  const float* dp = delta + (size_t)(b * DIMS + d) * SEQ + s0;
  float* yp = y + (size_t)(b * DIMS + d) * SEQ + s0;

  __builtin_prefetch(up + CHUNK, 0, 1);
  __builtin_prefetch(dp + CHUNK, 0, 1);

  for (int j = 0; j < CHUNK; j += 4) {
    float4 u4 = *(const float4*)(up + j);
    float4 l4 = *(const float4*)(dp + j);
    float4 y4;
#pragma unroll
    for (int k = 0; k < 4; k++) {
      float uk = ((const float*)&u4)[k];
      float dt = softplus_f(((const float*)&l4)[k] + bias);
      float du = dt * uk;
      const float* bj = &sB[j + k];
      const float* cj = &sC[j + k];
      float yv = Dd * uk;
#pragma unroll
      for (int n = 0; n < NST; n++) {
        float a = __expf(dt * Arow[n]);
        x[n] = a * x[n] + du * bj[n * CHUNK];
        yv += x[n] * cj[n * CHUNK];
      }
      ((float*)&y4)[k] = yv;
    }
    *(float4*)(yp + j) = y4;
  }
}

extern "C" void kernel_launch(void* const* d_in, const int* in_sizes, int n_in,
                              void* d_out, int out_size, void* d_ws, size_t ws_size,
                              hipStream_t stream) {
  const float* u     = (const float*)d_in[0];
  const float* delta = (const float*)d_in[1];
  const float* Bm    = (const float*)d_in[2];
  const float* Cm    = (const float*)d_in[3];
  const float* A     = (const float*)d_in[4];
  const float* Dv    = (const float*)d_in[5];
  const float* bias  = (const float*)d_in[6];
  float* y = (float*)d_out;

  float* aprod = (float*)d_ws;                                  // [B][NCH][D][N]
  float* xloc  = aprod + (size_t)BATCH * NCH * DIMS * NST;      // [B][NCH][D][N]

  dim3 grid(NDT, NCH, BATCH);
  ssm_chunk_local<<<grid, 256, 0, stream>>>(u, delta, Bm, A, bias, aprod, xloc);
  ssm_chunk_scan<<<(BATCH * DIMS * NST) / 256, 256, 0, stream>>>(aprod, xloc);
  ssm_chunk_out<<<grid, 256, 0, stream>>>(u, delta, Bm, Cm, A, Dv, bias, xloc, y);
}